// CoupledMamba_4063039062750
// MI455X (gfx1250) — compile-verified
//
#include <hip/hip_runtime.h>
#include <math.h>

typedef float v2f __attribute__((ext_vector_type(2)));
typedef float v8f __attribute__((ext_vector_type(8)));

#define WMMA_F32(a, b, c) \
    __builtin_amdgcn_wmma_f32_16x16x4_f32(false, (a), false, (b), (short)0, (c), false, false)

static constexpr int B_ = 32, S_ = 1024, D_ = 256, DS_ = 64, M_ = 3;
static constexpr int ROWS = B_ * S_;   // 32768 (b,s) rows per modality

// ---------------------------------------------------------------------------
// Kernel 1: input projection  u[m,row,c] = sum_k XW[m,c,k] * x[m,row,k] + Xb[m,c]
// One wave computes a 16-row x 64-col tile (4 f32 WMMA accumulators), K=256.
// ---------------------------------------------------------------------------
__global__ __launch_bounds__(256) void k1_inproj(
    const float* __restrict__ x0, const float* __restrict__ x1,
    const float* __restrict__ x2, const float* __restrict__ XW,
    const float* __restrict__ Xb, float* __restrict__ u)
{
    const int m    = blockIdx.y;
    const float* x = (m == 0) ? x0 : ((m == 1) ? x1 : x2);
    const int tid  = threadIdx.x;
    const int lane = tid & 31, wv = tid >> 5;
    const int r    = lane & 15, hi = lane >> 4;
    const int r0   = (blockIdx.x * 8 + wv) * 16;

    v8f acc0 = {}, acc1 = {}, acc2 = {}, acc3 = {};
    const float* arow = x + (size_t)(r0 + r) * D_;
    const float* bw0  = XW + ((size_t)m * DS_ +  0 + r) * D_;
    const float* bw1  = XW + ((size_t)m * DS_ + 16 + r) * D_;
    const float* bw2  = XW + ((size_t)m * DS_ + 32 + r) * D_;
    const float* bw3  = XW + ((size_t)m * DS_ + 48 + r) * D_;

    for (int k0 = 0; k0 < D_; k0 += 4) {
        const int kk = k0 + 2 * hi;
        float2 av = *(const float2*)(arow + kk);
        v2f a; a.x = av.x; a.y = av.y;
        float2 bv;
        v2f b;
        bv = *(const float2*)(bw0 + kk); b.x = bv.x; b.y = bv.y; acc0 = WMMA_F32(a, b, acc0);
        bv = *(const float2*)(bw1 + kk); b.x = bv.x; b.y = bv.y; acc1 = WMMA_F32(a, b, acc1);
        bv = *(const float2*)(bw2 + kk); b.x = bv.x; b.y = bv.y; acc2 = WMMA_F32(a, b, acc2);
        bv = *(const float2*)(bw3 + kk); b.x = bv.x; b.y = bv.y; acc3 = WMMA_F32(a, b, acc3);
    }

    // D element: row = r0 + j + 8*hi, col = ct*16 + (lane&15)
#pragma unroll
    for (int j = 0; j < 8; ++j) {
        const int row = r0 + j + 8 * hi;
        float* urow = u + ((size_t)m * ROWS + row) * DS_;
        urow[ 0 + r] = acc0[j] + Xb[m * DS_ +  0 + r];
        urow[16 + r] = acc1[j] + Xb[m * DS_ + 16 + r];
        urow[32 + r] = acc2[j] + Xb[m * DS_ + 32 + r];
        urow[48 + r] = acc3[j] + Xb[m * DS_ + 48 + r];
    }
}

// ---------------------------------------------------------------------------
// Kernel 2: sequential recurrence. 2 blocks x 16 batches, 256 threads (8 waves).
// CDNA5: 320 KB LDS per WGP -> stage ALL recurrence weights in LDS once
// (~254 KB incl. state/scratch), so the 1024-step hot loop touches only LDS
// (every WMMA fragment is a single ds_load_b64) plus one streamed u[t] read.
// ---------------------------------------------------------------------------
__global__ __launch_bounds__(256, 1) void k2_scan(
    const float* __restrict__ u, float* __restrict__ hseq,
    const float* __restrict__ HW, const float* __restrict__ Hb,
    const float* __restrict__ CW, const float* __restrict__ W1,
    const float* __restrict__ b1, const float* __restrict__ W2,
    const float* __restrict__ b2)
{
    // state + per-step scratch
    __shared__ float h_s [M_ * DS_ * 16];   // [m][k/2][bb][2]  12 KB (k-pair packed)
    __shared__ float su_s[M_ * DS_ * 16];   // [m][d][bb]       12 KB
    __shared__ float P_s [6  * DS_ * 16];   // [pair][d][bb]    24 KB
    __shared__ float z_s [16 * 66];         // [bb][e] pad-66    4.1 KB
    __shared__ float attn_s[16 * 12];       // [bb][t*3+s]
    // staged weights (row pads keep 8B alignment + even-bank spread)
    __shared__ float w1_s[DS_ * 194];       // [64][192->194]   48.5 KB
    __shared__ float hw_s[M_ * DS_ * 66];   // [3*64][64->66]   49.5 KB
    __shared__ float cw_s[6 * DS_ * 66];    // [6*64][64->66]   99.0 KB
    __shared__ float w2_s[16 * 66];         // [9->16][64->66]   4.1 KB

    const int tid  = threadIdx.x;
    const int lane = tid & 31, wv = tid >> 5;
    const int r    = lane & 15, hi = lane >> 4;
    const int b0   = blockIdx.x * 16;

    // ---- one-time staging ------------------------------------------------
    for (int i = tid; i < M_ * DS_ * 16; i += 256) h_s[i] = 0.0f;
    for (int i = tid; i < DS_ * (M_ * DS_); i += 256)        // W1 [64][192]
        w1_s[(i / 192) * 194 + (i % 192)] = W1[i];
    for (int i = tid; i < M_ * DS_ * DS_; i += 256)          // HW [3][64][64]
        hw_s[(i >> 6) * 66 + (i & 63)] = HW[i];
#pragma unroll
    for (int pr = 0; pr < 6; ++pr) {                         // CW off-diag pairs
        const int tt = pr >> 1;
        const int ss = (pr & 1) ? ((tt == 2) ? 1 : 2) : ((tt == 0) ? 1 : 0);
        const float* src = CW + (size_t)(tt * 3 + ss) * DS_ * DS_;
        for (int i = tid; i < DS_ * DS_; i += 256)
            cw_s[(pr * DS_ + (i >> 6)) * 66 + (i & 63)] = src[i];
    }
    for (int i = tid; i < 16 * 66; i += 256) w2_s[i] = 0.0f; // zero-pad rows 9..15
    __syncthreads();
    for (int i = tid; i < 9 * DS_; i += 256)                 // W2 [9][64]
        w2_s[(i >> 6) * 66 + (i & 63)] = W2[i];
    __syncthreads();

    for (int t = 0; t < S_; ++t) {
        // Warm L2/WGP$ with next step's u (global_prefetch_b8)
        if (t + 1 < S_ && tid < 96) {
            const int mm = tid >> 5, rem = tid & 31;
            const int bb = rem >> 1, half = (rem & 1) * 32;
            __builtin_prefetch(
                u + ((size_t)mm * ROWS + (size_t)(b0 + bb) * S_ + (t + 1)) * DS_ + half, 0, 3);
        }

        // ---- Phase A: 40 WMMA tile tasks spread across 8 waves -------------
        for (int task = wv; task < 40; task += 8) {
            v8f acc = {};
            if (task < 4) {
                // MLP layer 1: z[bb, ct*16+col] over K = M*DS = 192
                const int ct = task;
                const float* w1r = w1_s + (ct * 16 + r) * 194;
                for (int k0 = 0; k0 < M_ * DS_; k0 += 4) {
                    const int kk = k0 + 2 * hi;
                    const int ms = kk >> 6, ko = kk & 63;
                    float2 av = *(const float2*)(h_s + ms * 1024 + (ko >> 1) * 32 + r * 2);
                    v2f a; a.x = av.x; a.y = av.y;
                    float2 bv = *(const float2*)(w1r + kk);
                    v2f b; b.x = bv.x; b.y = bv.y;
                    acc = WMMA_F32(a, b, acc);
                }
#pragma unroll
                for (int j = 0; j < 8; ++j) {
                    const int bb = j + 8 * hi, col = ct * 16 + r;
                    z_s[bb * 66 + col] = tanhf(acc[j] + b1[col]);
                }
            } else if (task < 16) {
                // su_lin[m] = HW[m] @ h[m]   (d rows x batch cols, K = 64)
                const int idx = task - 4, mm = idx >> 2, ct = idx & 3;
                const float* ar = hw_s + (mm * DS_ + ct * 16 + r) * 66;
                const float* hb = h_s + mm * 1024;
                for (int k0 = 0; k0 < DS_; k0 += 4) {
                    const int kk = k0 + 2 * hi;
                    float2 av = *(const float2*)(ar + kk);
                    v2f a; a.x = av.x; a.y = av.y;
                    float2 hv = *(const float2*)(hb + (kk >> 1) * 32 + r * 2);
                    v2f b; b.x = hv.x; b.y = hv.y;
                    acc = WMMA_F32(a, b, acc);
                }
#pragma unroll
                for (int j = 0; j < 8; ++j)
                    su_s[(mm * DS_ + ct * 16 + j + 8 * hi) * 16 + r] = acc[j];
            } else {
                // P[pr] = CW[tt,ss] @ h[ss]   (6 off-diagonal pairs)
                const int idx = task - 16, pr = idx >> 2, ct = idx & 3;
                const int tt = pr >> 1;
                const int ss = (pr & 1) ? ((tt == 2) ? 1 : 2) : ((tt == 0) ? 1 : 0);
                const float* ar = cw_s + (pr * DS_ + ct * 16 + r) * 66;
                const float* hb = h_s + ss * 1024;
                for (int k0 = 0; k0 < DS_; k0 += 4) {
                    const int kk = k0 + 2 * hi;
                    float2 av = *(const float2*)(ar + kk);
                    v2f a; a.x = av.x; a.y = av.y;
                    float2 hv = *(const float2*)(hb + (kk >> 1) * 32 + r * 2);
                    v2f b; b.x = hv.x; b.y = hv.y;
                    acc = WMMA_F32(a, b, acc);
                }
#pragma unroll
                for (int j = 0; j < 8; ++j)
                    P_s[(pr * DS_ + ct * 16 + j + 8 * hi) * 16 + r] = acc[j];
            }
        }
        __syncthreads();

        // ---- MLP layer 2 (wave 0): raw[bb, o<9] = z @ W2.T + b2 ------------
        if (wv == 0) {
            v8f acc = {};
            for (int k0 = 0; k0 < DS_; k0 += 4) {
                const int kk = k0 + 2 * hi;
                float2 av = *(const float2*)(z_s + r * 66 + kk);
                v2f a; a.x = av.x; a.y = av.y;
                float2 bv = *(const float2*)(w2_s + r * 66 + kk);  // rows>=9 are zero
                v2f b; b.x = bv.x; b.y = bv.y;
                acc = WMMA_F32(a, b, acc);
            }
#pragma unroll
            for (int j = 0; j < 8; ++j) {
                const int bb = j + 8 * hi;
                if (r < 9) attn_s[bb * 12 + r] = acc[j] + b2[r];
            }
        }
        __syncthreads();

        // ---- softmax over src (3) per (bb, tgt) ----------------------------
        if (tid < 48) {
            const int bb = tid / 3, tg = tid % 3;
            float v0 = attn_s[bb * 12 + tg * 3 + 0];
            float v1 = attn_s[bb * 12 + tg * 3 + 1];
            float v2 = attn_s[bb * 12 + tg * 3 + 2];
            float mx = fmaxf(v0, fmaxf(v1, v2));
            float e0 = __expf(v0 - mx), e1 = __expf(v1 - mx), e2 = __expf(v2 - mx);
            float inv = 1.0f / (e0 + e1 + e2);
            attn_s[bb * 12 + tg * 3 + 0] = e0 * inv;
            attn_s[bb * 12 + tg * 3 + 1] = e1 * inv;
            attn_s[bb * 12 + tg * 3 + 2] = e2 * inv;
        }
        __syncthreads();

        // ---- combine: su = su_lin + Hb + u[t] + masked attn * P; SiLU ------
#pragma unroll
        for (int i = 0; i < 12; ++i) {
            const int idx = tid + i * 256;        // idx = (m*16 + bb)*64 + d
            const int d  = idx & 63;
            const int bb = (idx >> 6) & 15;
            const int m  = idx >> 10;
            float su = su_s[(m * DS_ + d) * 16 + bb] + Hb[m * DS_ + d]
                     + u[((size_t)m * ROWS + (size_t)(b0 + bb) * S_ + t) * DS_ + d];
#pragma unroll
            for (int s = 0; s < M_; ++s) {
                if (s == m) continue;
                const int pr = m * 2 + (s > m ? s - 1 : s);
                su += attn_s[bb * 12 + m * 3 + s] * P_s[(pr * DS_ + d) * 16 + bb];
            }
            const float hn = su / (1.0f + __expf(-su));  // SiLU
            h_s[m * 1024 + (d >> 1) * 32 + bb * 2 + (d & 1)] = hn;   // k-pair packed
            hseq[((size_t)m * ROWS + (size_t)(b0 + bb) * S_ + t) * DS_ + d] = hn;
        }
        __syncthreads();
    }
}

// ---------------------------------------------------------------------------
// Kernel 3: y = OW[m] @ h + Ob, out = LN(y + x) * g + b.
// One wave: 16-row x 256-col tile (16 WMMA accumulators, K=64).
// LayerNorm stats in-register via 16-lane __shfl_xor tree (rows split by half).
// ---------------------------------------------------------------------------
__global__ __launch_bounds__(256) void k3_outproj(
    const float* __restrict__ hseq, const float* __restrict__ OW,
    const float* __restrict__ Ob,
    const float* __restrict__ x0, const float* __restrict__ x1,
    const float* __restrict__ x2,
    const float* __restrict__ ln_g, const float* __restrict__ ln_b,
    float* __restrict__ out)
{
    const int m    = blockIdx.y;
    const float* x = (m == 0) ? x0 : ((m == 1) ? x1 : x2);
    const int tid  = threadIdx.x;
    const int lane = tid & 31, wv = tid >> 5;
    const int r    = lane & 15, hi = lane >> 4;
    const int r0   = (blockIdx.x * 8 + wv) * 16;

    v8f acc[16];
#pragma unroll
    for (int ct = 0; ct < 16; ++ct) acc[ct] = (v8f){};

    const float* arow = hseq + ((size_t)m * ROWS + r0 + r) * DS_;
    for (int k0 = 0; k0 < DS_; k0 += 4) {
        const int kk = k0 + 2 * hi;
        float2 av = *(const float2*)(arow + kk);
        v2f a; a.x = av.x; a.y = av.y;
#pragma unroll
        for (int ct = 0; ct < 16; ++ct) {
            float2 bv = *(const float2*)(OW + ((size_t)m * D_ + ct * 16 + r) * DS_ + kk);
            v2f b; b.x = bv.x; b.y = bv.y;
            acc[ct] = WMMA_F32(a, b, acc[ct]);
        }
    }

    // residual + row statistics (row j lives in this lane's half)
    float s1[8], s2[8];
#pragma unroll
    for (int j = 0; j < 8; ++j) { s1[j] = 0.f; s2[j] = 0.f; }
#pragma unroll
    for (int ct = 0; ct < 16; ++ct) {
#pragma unroll
        for (int j = 0; j < 8; ++j) {
            const int row = r0 + j + 8 * hi;
            const int col = ct * 16 + r;
            float v = acc[ct][j] + Ob[m * D_ + col] + x[(size_t)row * D_ + col];
            acc[ct][j] = v;
            s1[j] += v;
            s2[j] += v * v;
        }
    }
#pragma unroll
    for (int j = 0; j < 8; ++j) {
#pragma unroll
        for (int off = 1; off <= 8; off <<= 1) {   // stays within 16-lane half
            s1[j] += __shfl_xor(s1[j], off, 32);
            s2[j] += __shfl_xor(s2[j], off, 32);
        }
    }
    float mu[8], sc[8];
#pragma unroll
    for (int j = 0; j < 8; ++j) {
        mu[j] = s1[j] * (1.0f / D_);
        float var = s2[j] * (1.0f / D_) - mu[j] * mu[j];
        sc[j] = rsqrtf(var + 1e-5f);
    }
#pragma unroll
    for (int ct = 0; ct < 16; ++ct) {
#pragma unroll
        for (int j = 0; j < 8; ++j) {
            const int row = r0 + j + 8 * hi;
            const int col = ct * 16 + r;
            out[((size_t)m * ROWS + row) * D_ + col] =
                (acc[ct][j] - mu[j]) * sc[j] * ln_g[m * D_ + col] + ln_b[m * D_ + col];
        }
    }
}

// ---------------------------------------------------------------------------
extern "C" void kernel_launch(void* const* d_in, const int* in_sizes, int n_in,
                              void* d_out, int out_size, void* d_ws, size_t ws_size,
                              hipStream_t stream)
{
    (void)in_sizes; (void)n_in; (void)out_size; (void)ws_size;
    const float* x0   = (const float*)d_in[0];
    const float* x1   = (const float*)d_in[1];
    const float* x2   = (const float*)d_in[2];
    const float* XW   = (const float*)d_in[3];
    const float* Xb   = (const float*)d_in[4];
    const float* HW   = (const float*)d_in[5];
    const float* Hb   = (const float*)d_in[6];
    const float* OW   = (const float*)d_in[7];
    const float* Ob   = (const float*)d_in[8];
    const float* CW   = (const float*)d_in[9];
    const float* W1   = (const float*)d_in[10];
    const float* b1   = (const float*)d_in[11];
    const float* W2   = (const float*)d_in[12];
    const float* b2   = (const float*)d_in[13];
    const float* ln_g = (const float*)d_in[14];
    const float* ln_b = (const float*)d_in[15];

    float* u    = (float*)d_ws;                         // [M][B*S][DS]  25.2 MB
    float* hseq = u + (size_t)M_ * ROWS * DS_;          // [M][B*S][DS]  25.2 MB

    dim3 g1(ROWS / 16 / 8, M_);   // 256 x 3
    k1_inproj<<<g1, 256, 0, stream>>>(x0, x1, x2, XW, Xb, u);

    k2_scan<<<2, 256, 0, stream>>>(u, hseq, HW, Hb, CW, W1, b1, W2, b2);

    dim3 g3(ROWS / 16 / 8, M_);   // 256 x 3
    k3_outproj<<<g3, 256, 0, stream>>>(hseq, OW, Ob, x0, x1, x2, ln_g, ln_b,
                                       (float*)d_out);
}